// HardVoxelization_55834574848166
// MI455X (gfx1250) — compile-verified
//
#include <hip/hip_runtime.h>
#include <stdint.h>

// ---------------- problem constants (match reference) ----------------
#define GXD 1408
#define GYD 1600
#define GZD 40
#define GYZ (GYD * GZD)            // 64000
#define BIGID (GXD * GYD * GZD)    // 90,112,000  (< 2^27)
#define NPTS_MAX 2000000
#define MAX_VOX 40000
#define MAX_PTS 32
#define BITWORDS (BIGID / 32)      // 2,816,000 (exact)
#define POPC_BLK 256
#define NPOPC_BLOCKS (BITWORDS / POPC_BLK)  // 11,000 (exact)
#define EMPTY_SLOT 0x7fffffff

// ---------------- K0: reset workspace each call (graph-replay safe) ----
__global__ void hv_init_ws(unsigned* __restrict__ bitmap,
                           int* __restrict__ uniq,
                           int* __restrict__ slots) {
    int i = blockIdx.x * blockDim.x + threadIdx.x;
    if (i < BITWORDS) bitmap[i] = 0u;
    if (i < MAX_VOX) uniq[i] = BIGID;
    if (i < MAX_VOX * MAX_PTS) slots[i] = EMPTY_SLOT;
}

// ---------------- K1: per-point linear voxel id + presence bitmap ------
__global__ void hv_compute_lin(const float4* __restrict__ points,
                               int* __restrict__ lin,
                               unsigned* __restrict__ bitmap, int n) {
    int i = blockIdx.x * blockDim.x + threadIdx.x;
    if (i >= n) return;
    float4 p = points[i];
    // (p - pcr_lo) / voxel_size, floored -- identical IEEE f32 ops to JAX
    float fx = floorf((p.x - 0.0f)    / 0.05f);
    float fy = floorf((p.y - (-40.0f))/ 0.05f);
    float fz = floorf((p.z - (-3.0f)) / 0.1f);
    int vx = (int)fx, vy = (int)fy, vz = (int)fz;
    bool ok = (vx >= 0) & (vx < GXD) & (vy >= 0) & (vy < GYD) &
              (vz >= 0) & (vz < GZD);
    int l = ok ? (vx * GYZ + vy * GZD + vz) : BIGID;
    lin[i] = l;
    if (ok) atomicOr(&bitmap[(unsigned)l >> 5], 1u << (l & 31));
}

// ---------------- K2: per-block popcount of bitmap ---------------------
__global__ void hv_popc(const unsigned* __restrict__ bitmap,
                        int* __restrict__ blockSums) {
    __shared__ int sm[POPC_BLK];
    int tid = threadIdx.x;
    int gid = blockIdx.x * POPC_BLK + tid;   // BITWORDS is exact multiple
    sm[tid] = __popc(bitmap[gid]);
    __syncthreads();
    for (int s = POPC_BLK / 2; s > 0; s >>= 1) {
        if (tid < s) sm[tid] += sm[tid + s];
        __syncthreads();
    }
    if (tid == 0) blockSums[blockIdx.x] = sm[0];
}

// ---------------- K3: exclusive scan of block sums (one block) ---------
__global__ void hv_scan(const int* __restrict__ blockSums,
                        int* __restrict__ blockOffs) {
    __shared__ int sm[1024];
    __shared__ int carry;
    int tid = threadIdx.x;
    if (tid == 0) carry = 0;
    __syncthreads();
    for (int base = 0; base < NPOPC_BLOCKS; base += 1024) {
        int idx = base + tid;
        int v = (idx < NPOPC_BLOCKS) ? blockSums[idx] : 0;
        sm[tid] = v;
        __syncthreads();
        for (int off = 1; off < 1024; off <<= 1) {
            int t = (tid >= off) ? sm[tid - off] : 0;
            __syncthreads();
            sm[tid] += t;
            __syncthreads();
        }
        int c = carry;
        if (idx < NPOPC_BLOCKS) blockOffs[idx] = c + sm[tid] - v;
        __syncthreads();                 // all reads of carry done
        if (tid == 0) carry = c + sm[1023];
        __syncthreads();
    }
}

// ---------------- K4: emit the MAX_VOX smallest unique ids -------------
__global__ void hv_emit_uniq(const unsigned* __restrict__ bitmap,
                             const int* __restrict__ blockOffs,
                             int* __restrict__ uniq) {
    __shared__ int sm[POPC_BLK];
    int tid = threadIdx.x;
    int gid = blockIdx.x * POPC_BLK + tid;
    unsigned w = bitmap[gid];
    int c = __popc(w);
    sm[tid] = c;
    __syncthreads();
    for (int off = 1; off < POPC_BLK; off <<= 1) {
        int t = (tid >= off) ? sm[tid - off] : 0;
        __syncthreads();
        sm[tid] += t;
        __syncthreads();
    }
    int r = blockOffs[blockIdx.x] + sm[tid] - c;  // global rank of first bit
    if (r >= MAX_VOX) return;
    unsigned ww = w;
    int base = gid << 5;
    while (ww) {
        int b = __ffs(ww) - 1;            // bits LSB->MSB == increasing id
        if (r >= MAX_VOX) break;
        uniq[r] = base + b;
        ww &= ww - 1;
        r++;
    }
}

// ---------------- K5: rank-preserving insertion (min-32 by index) ------
__global__ void hv_insert(const int* __restrict__ lin,
                          const int* __restrict__ uniq,
                          int* __restrict__ slots, int n) {
    int i = blockIdx.x * blockDim.x + threadIdx.x;
    if (i >= n) return;
    int l = lin[i];
    if (l >= BIGID) return;
    // binary search in sorted uniq[0..MAX_VOX)
    int lo = 0, hi = MAX_VOX;
    while (lo < hi) {
        int mid = (lo + hi) >> 1;
        if (uniq[mid] < l) lo = mid + 1; else hi = mid;
    }
    if (lo >= MAX_VOX || uniq[lo] != l) return;   // voxel beyond 40000 smallest
    int* slot = slots + lo * MAX_PTS;
    int key = i;
    #pragma unroll 4
    for (int s = 0; s < MAX_PTS; ++s) {
        int old = atomicMin(&slot[s], key);
        if (old == EMPTY_SLOT) return;   // placed in empty slot
        if (old > key) key = old;        // displaced a larger index, carry it
        // else our key is still the larger one; try next slot
    }
    // fell off the end: rank >= 32, dropped (matches reference)
}

// ---------------- K6: one wave32 per voxel -> outputs ------------------
__global__ void hv_finalize(const float4* __restrict__ points,
                            const int* __restrict__ slots,
                            const int* __restrict__ uniq,
                            float4* __restrict__ voxOut,
                            int* __restrict__ coordsOut,
                            int* __restrict__ nppvOut) {
    int v = blockIdx.x * 8 + (threadIdx.x >> 5);  // 8 voxels / 256-thr block
    int lane = threadIdx.x & 31;                  // wave32: lane == slot
    int idx = slots[v * MAX_PTS + lane];
    bool has = (idx != EMPTY_SLOT);
    float4 out = make_float4(0.f, 0.f, 0.f, 0.f);
    if (has) out = points[idx];
    voxOut[v * MAX_PTS + lane] = out;             // rewrites whole region
    unsigned long long m = __ballot(has);         // wave32: low 32 bits
    if (lane == 0) {
        nppvOut[v] = __popcll(m);
        int u = uniq[v];
        int cx = 0, cy = 0, cz = 0;
        if (u < BIGID) {
            cx = u / GYZ;
            int rem = u - cx * GYZ;
            cy = rem / GZD;
            cz = rem - cy * GZD;
        }
        coordsOut[v * 3 + 0] = cx;
        coordsOut[v * 3 + 1] = cy;
        coordsOut[v * 3 + 2] = cz;
    }
}

// ---------------- host-side launch -------------------------------------
extern "C" void kernel_launch(void* const* d_in, const int* in_sizes, int n_in,
                              void* d_out, int out_size, void* d_ws, size_t ws_size,
                              hipStream_t stream) {
    const float4* points = (const float4*)d_in[0];
    int n = in_sizes[0] / 4;                 // 2,000,000 points

    // workspace layout (ints): lin | bitmap | blockSums | blockOffs | uniq | slots
    int* ws = (int*)d_ws;
    int* lin            = ws;
    unsigned* bitmap    = (unsigned*)(lin + NPTS_MAX);
    int* blockSums      = (int*)(bitmap + BITWORDS);
    int* blockOffs      = blockSums + NPOPC_BLOCKS;
    int* uniq           = blockOffs + NPOPC_BLOCKS;
    int* slots          = uniq + MAX_VOX;

    // output layout: voxels f32 | coords i32 | nppv i32 (flat, in return order)
    float4* voxOut = (float4*)d_out;
    int* coordsOut = (int*)d_out + MAX_VOX * MAX_PTS * 4;
    int* nppvOut   = coordsOut + MAX_VOX * 3;

    hv_init_ws<<<(BITWORDS + 255) / 256, 256, 0, stream>>>(bitmap, uniq, slots);
    hv_compute_lin<<<(n + 255) / 256, 256, 0, stream>>>(points, lin, bitmap, n);
    hv_popc<<<NPOPC_BLOCKS, POPC_BLK, 0, stream>>>(bitmap, blockSums);
    hv_scan<<<1, 1024, 0, stream>>>(blockSums, blockOffs);
    hv_emit_uniq<<<NPOPC_BLOCKS, POPC_BLK, 0, stream>>>(bitmap, blockOffs, uniq);
    hv_insert<<<(n + 255) / 256, 256, 0, stream>>>(lin, uniq, slots, n);
    hv_finalize<<<MAX_VOX / 8, 256, 0, stream>>>(points, slots, uniq,
                                                 voxOut, coordsOut, nppvOut);
}